// StableFANSAN_14353780704086
// MI455X (gfx1250) — compile-verified
//
#include <hip/hip_runtime.h>
#include <hip/hip_bf16.h>

typedef __attribute__((ext_vector_type(2))) float v2f;
typedef __attribute__((ext_vector_type(8))) float v8f;
typedef __attribute__((ext_vector_type(4))) float f4;

#define BDIM 64
#define TDIM 2048
#define HDIM 256
#define TSEG 256
#define NSEG 8             // TDIM / TSEG
#define NROW (BDIM * NSEG) // 512 partial rows, ordered seg = tseg*64 + b
#define EPSF 1e-4f

__device__ __forceinline__ void edge_stat(float s1, float s2, float w,
                                          float& mu, float& isd) {
  mu = s1 / w;
  float var = fmaxf((s2 - s1 * s1 / w) / (w - 1.0f), 0.0f);
  // max(sqrt(var), EPS) == sqrt(max(var, EPS^2)); keep reciprocal directly
  isd = __frsqrt_rn(fmaxf(var, EPSF * EPSF));
}

// ---------------------------------------------------------------------------
// Pass 1: streaming FAN along T with a 20-deep register ring buffer.
// grid = 512 blocks (seg = tseg*64 + b), block = 256 threads (thread <-> h).
// x is read non-temporal (pure stream); fan is stored with TH_STORE_WB so the
// 134MB intermediate stays dirty-resident in the 192MB L2 for pass 3.
// ---------------------------------------------------------------------------
__global__ __launch_bounds__(256) void fan_pass1(
    const float* __restrict__ x, const float* __restrict__ alpha,
    float* __restrict__ fan, float* __restrict__ psum, float* __restrict__ psq) {
  const int bx = blockIdx.x;
  const int tseg = bx >> 6;
  const int b = bx & 63;
  const int h = threadIdx.x;
  const int t0 = tseg * TSEG;
  const float* xp = x + (size_t)b * TDIM * HDIM + h;  // index with t*HDIM

  // per-h softmax(alpha[h, 0..2])
  float a0 = alpha[h * 3 + 0], a1 = alpha[h * 3 + 1], a2 = alpha[h * 3 + 2];
  float mx = fmaxf(a0, fmaxf(a1, a2));
  float e0 = __expf(a0 - mx), e1 = __expf(a1 - mx), e2 = __expf(a2 - mx);
  float inv = 1.0f / (e0 + e1 + e2);
  float al0 = e0 * inv, al1 = e1 * inv, al2 = e2 * inv;

  float ring[20];
  float s1_5 = 0.f, s1_10 = 0.f, s1_20 = 0.f;
  float s2_5 = 0.f, s2_10 = 0.f, s2_20 = 0.f;
  float mu_e0 = 0.f, mu_e1 = 0.f, mu_e2 = 0.f;
  float is_e0 = 0.f, is_e1 = 0.f, is_e2 = 0.f;

  if (tseg == 0) {
#pragma unroll
    for (int p = 0; p < 20; ++p) ring[p] = 0.0f;
    // edge (pad-mode 'edge') stats from the first full window of each size
    float s1 = 0.f, s2 = 0.f;
#pragma unroll
    for (int p = 0; p < 20; ++p) {
      float v = __builtin_nontemporal_load(&xp[(size_t)p * HDIM]);
      s1 += v;
      s2 += v * v;
      if (p == 4)  edge_stat(s1, s2, 5.0f,  mu_e0, is_e0);
      if (p == 9)  edge_stat(s1, s2, 10.0f, mu_e1, is_e1);
      if (p == 19) edge_stat(s1, s2, 20.0f, mu_e2, is_e2);
    }
  } else {
    // halo: prefill ring with x[t0-20 .. t0-1]; ring[p] = x[t0-20+p]
#pragma unroll
    for (int p = 0; p < 20; ++p)
      ring[p] = __builtin_nontemporal_load(&xp[(size_t)(t0 - 20 + p) * HDIM]);
#pragma unroll
    for (int p = 0; p < 20; ++p) {
      float v = ring[p], v2 = v * v;
      s1_20 += v; s2_20 += v2;
      if (p >= 10) { s1_10 += v; s2_10 += v2; }
      if (p >= 15) { s1_5  += v; s2_5  += v2; }
    }
  }

  float ssum = 0.f, ssq = 0.f;
  const float* xq = xp + (size_t)t0 * HDIM;
  // byte offset of fan[b][t0][h] (fits in 32 bits: <= 128MiB)
  const unsigned fbase = ((unsigned)(b * TDIM + t0) * HDIM + (unsigned)h) * 4u;

#pragma unroll 1
  for (int j = 0; j < 13; ++j) {
#pragma unroll
    for (int u = 0; u < 20; ++u) {
      const int i = j * 20 + u;
      if (i < TSEG) {
        float xt = __builtin_nontemporal_load(&xq[(size_t)i * HDIM]);
        float o20 = ring[u];
        float o10 = ring[(u + 10) % 20];
        float o5  = ring[(u + 15) % 20];
        float xs = xt * xt;
        s1_20 += xt - o20; s2_20 += xs - o20 * o20;
        s1_10 += xt - o10; s2_10 += xs - o10 * o10;
        s1_5  += xt - o5;  s2_5  += xs - o5 * o5;
        ring[u] = xt;
        const int t = t0 + i;

        float mu5, is5;
        {
          float m = s1_5 * 0.2f;
          float var = fmaxf((s2_5 - s1_5 * s1_5 * 0.2f) * 0.25f, 0.0f);
          float is = __frsqrt_rn(fmaxf(var, EPSF * EPSF));
          mu5 = (t >= 4) ? m : mu_e0;  is5 = (t >= 4) ? is : is_e0;
        }
        float mu10, is10;
        {
          float m = s1_10 * 0.1f;
          float var = fmaxf((s2_10 - s1_10 * s1_10 * 0.1f) * (1.0f / 9.0f), 0.0f);
          float is = __frsqrt_rn(fmaxf(var, EPSF * EPSF));
          mu10 = (t >= 9) ? m : mu_e1;  is10 = (t >= 9) ? is : is_e1;
        }
        float mu20, is20;
        {
          float m = s1_20 * 0.05f;
          float var = fmaxf((s2_20 - s1_20 * s1_20 * 0.05f) * (1.0f / 19.0f), 0.0f);
          float is = __frsqrt_rn(fmaxf(var, EPSF * EPSF));
          mu20 = (t >= 19) ? m : mu_e2;  is20 = (t >= 19) ? is : is_e2;
        }
        float acc = (xt - mu5)  * is5  * al0 +
                    (xt - mu10) * is10 * al1 +
                    (xt - mu20) * is20 * al2;
        // keep fan dirty-resident in far cache (L2), overriding write-rinse
        {
          unsigned foff = fbase + (unsigned)i * (HDIM * 4u);
          asm volatile(
              "global_store_b32 %0, %1, %2 th:TH_STORE_WB scope:SCOPE_DEV"
              :: "v"(foff), "v"(acc), "s"(fan));
        }
        ssum += acc;
        ssq += acc * acc;
      }
    }
  }
  psum[bx * HDIM + h] = ssum;
  psq[bx * HDIM + h] = ssq;
}

// ---------------------------------------------------------------------------
// Pass 2: slice statistics via V_WMMA_F32_16X16X4_F32 column-sum reduction.
// 512 partial rows reduced with 0/1 weights in A (slice membership, uniform
// within each K-chunk of 4 because rows are ordered tseg-major). grid = 32
// one-wave blocks: blockIdx -> (slice k, 16-column tile).
// ---------------------------------------------------------------------------
__global__ __launch_bounds__(32) void san_stats_wmma(
    const float* __restrict__ psum, const float* __restrict__ psq,
    float* __restrict__ mu, float* __restrict__ sd) {
  const int k = blockIdx.x >> 4;
  const int ct = blockIdx.x & 15;
  const int l = threadIdx.x;
  const int col = ct * 16 + (l & 15);
  const int ro = (l < 16) ? 0 : 2;

  v8f cs = {};
  v8f cq = {};
#pragma unroll 4
  for (int kk = 0; kk < NROW / 4; ++kk) {
    const int base = kk * 4;
    const int tseg = base >> 6;  // uniform across the 4-row chunk
    const float w = (k == 0) ? ((tseg <= 5) ? 1.0f : 0.0f)
                             : ((tseg >= 4) ? 1.0f : 0.0f);
    v2f a; a.x = w; a.y = w;                 // A[m][k] = w for all m,k
    v2f bs, bq;                              // 4 rows across (vgpr, lane-half)
    bs.x = psum[(base + ro) * HDIM + col];
    bs.y = psum[(base + ro + 1) * HDIM + col];
    bq.x = psq[(base + ro) * HDIM + col];
    bq.y = psq[(base + ro + 1) * HDIM + col];
    cs = __builtin_amdgcn_wmma_f32_16x16x4_f32(false, a, false, bs, (short)0,
                                               cs, false, false);
    cq = __builtin_amdgcn_wmma_f32_16x16x4_f32(false, a, false, bq, (short)0,
                                               cq, false, false);
  }
  // D rows are identical; lane n (n = lane%16) holds the sum for column n.
  float S = cs[0];
  float Q = cq[0];
  float cnt = (k == 0) ? (float)(BDIM * 1536) : (float)(BDIM * 1024);
  float m = S / cnt;
  float var = fmaxf((Q - S * S / cnt) / (cnt - 1.0f), 0.0f);
  float s = fmaxf(sqrtf(var), EPSF);
  if (l < 16) {
    mu[k * HDIM + col] = m;
    sd[k * HDIM + col] = s;
  }
}

// ---------------------------------------------------------------------------
// Pass 2b: fold mu/sd/gamma/beta/overlap-count into per-region affine coeffs:
// region 0: t in [0,1024) (slice 0), 1: [1024,1536) (both), 2: [1536,2048).
// ---------------------------------------------------------------------------
__global__ __launch_bounds__(256) void san_coeff(
    const float* __restrict__ mu, const float* __restrict__ sd,
    const float* __restrict__ gamma, const float* __restrict__ beta,
    float* __restrict__ ca, float* __restrict__ cb) {
  const int h = threadIdx.x;
  float g0 = gamma[0 * HDIM + h], g1 = gamma[1 * HDIM + h];
  float b0 = beta[0 * HDIM + h],  b1 = beta[1 * HDIM + h];
  float m0 = mu[0 * HDIM + h], m1 = mu[1 * HDIM + h];
  float s0 = sd[0 * HDIM + h], s1 = sd[1 * HDIM + h];
  float a0 = g0 / s0, a1 = g1 / s1;
  float c0 = b0 - a0 * m0, c1 = b1 - a1 * m1;
  float i1 = 1.0f / (1.0f + EPSF), i2 = 1.0f / (2.0f + EPSF);
  ca[0 * HDIM + h] = a0 * i1;        cb[0 * HDIM + h] = c0 * i1;
  ca[1 * HDIM + h] = (a0 + a1) * i2; cb[1 * HDIM + h] = (c0 + c1) * i2;
  ca[2 * HDIM + h] = a1 * i1;        cb[2 * HDIM + h] = c1 * i1;
}

// ---------------------------------------------------------------------------
// Pass 3: out = x + fan * ca[r][h] + cb[r][h], float4-vectorized streaming.
// fan is read with TH_LOAD_LU (last-use): hits the dirty L2 line from pass 1
// and discards it — the 134MB intermediate ideally never touches HBM.
// ---------------------------------------------------------------------------
__global__ __launch_bounds__(256) void final_pass(
    const float* __restrict__ x, const float* __restrict__ fan,
    const float* __restrict__ ca, const float* __restrict__ cb,
    float* __restrict__ out) {
  const size_t gid = (size_t)blockIdx.x * blockDim.x + threadIdx.x;  // float4
  const size_t row = gid >> 6;  // 64 float4 per (b,t) row
  const int h = (int)(gid & 63) * 4;
  const int t = (int)(row & (TDIM - 1));
  const int r = (t < 1024) ? 0 : ((t < 1536) ? 1 : 2);

  const f4 xv = __builtin_nontemporal_load((const f4*)(x + row * HDIM + h));
  f4 fv;
  {
    const unsigned foff = (unsigned)(gid * 16u);  // byte offset, < 2^27
    asm volatile("global_load_b128 %0, %1, %2 th:TH_LOAD_LU"
                 : "=v"(fv)
                 : "v"(foff), "s"(fan));
    // glue the wait to fv so no use can be scheduled before it
    asm volatile("s_wait_loadcnt 0x0" : "+v"(fv));
  }
  const f4 av = *(const f4*)(ca + r * HDIM + h);
  const f4 bv = *(const f4*)(cb + r * HDIM + h);
  f4 o;
  o.x = xv.x + fv.x * av.x + bv.x;
  o.y = xv.y + fv.y * av.y + bv.y;
  o.z = xv.z + fv.z * av.z + bv.z;
  o.w = xv.w + fv.w * av.w + bv.w;
  __builtin_nontemporal_store(o, (f4*)(out + row * HDIM + h));
}

extern "C" void kernel_launch(void* const* d_in, const int* in_sizes, int n_in,
                              void* d_out, int out_size, void* d_ws,
                              size_t ws_size, hipStream_t stream) {
  const float* x = (const float*)d_in[0];      // (64, 2048, 256)
  const float* alpha = (const float*)d_in[1];  // (256, 3)
  const float* gamma = (const float*)d_in[2];  // (3, 256)
  const float* beta = (const float*)d_in[3];   // (3, 256)
  float* out = (float*)d_out;

  float* fan = (float*)d_ws;
  const size_t F = (size_t)BDIM * TDIM * HDIM;
  float* psum = fan + F;
  float* psq = psum + (size_t)NROW * HDIM;
  float* mu = psq + (size_t)NROW * HDIM;
  float* sd = mu + 2 * HDIM;
  float* ca = sd + 2 * HDIM;
  float* cb = ca + 3 * HDIM;

  fan_pass1<<<NROW, 256, 0, stream>>>(x, alpha, fan, psum, psq);
  san_stats_wmma<<<32, 32, 0, stream>>>(psum, psq, mu, sd);
  san_coeff<<<1, 256, 0, stream>>>(mu, sd, gamma, beta, ca, cb);
  final_pass<<<(int)(F / 4 / 256), 256, 0, stream>>>(x, fan, ca, cb, out);
}